// NearestGridSampler_88837103551029
// MI455X (gfx1250) — compile-verified
//
#include <hip/hip_runtime.h>
#include <stdint.h>

// ---------------- problem constants (from reference) ----------------
#define RES      128
#define NCELLS   (RES * RES * RES)        // 2,097,152 cells (8 MB fp32, L2-resident)
#define NTILES   1024
#define TILEC    (NCELLS / NTILES)        // 2048 cells per tile
#define ORIGINF  (-1.0f)                  // CENTER - SCALE
#define INV_CELL 64.0f                    // 1 / (2*SCALE/RES)
#define CELLSZ   0.015625f                // 2*SCALE/RES

typedef __attribute__((ext_vector_type(2))) float v2f;
typedef __attribute__((ext_vector_type(8))) float v8f;
typedef int v4i __attribute__((vector_size(16)));   // matches builtin prototype

// ---------------- gfx1250 feature probes ----------------
#if defined(__has_builtin)
# if __has_builtin(__builtin_amdgcn_wmma_f32_16x16x4_f32)
#  define HAVE_WMMA_F32X4 1
# endif
# if __has_builtin(__builtin_amdgcn_global_load_async_to_lds_b128) && \
     __has_builtin(__builtin_amdgcn_s_wait_asynccnt)
#  define HAVE_ASYNC_LDS 1
# endif
#endif

__device__ __forceinline__ int voxel_index(float x, float y, float z) {
  int i = (int)floorf((x - ORIGINF) * INV_CELL);
  int j = (int)floorf((y - ORIGINF) * INV_CELL);
  int k = (int)floorf((z - ORIGINF) * INV_CELL);
  i = min(RES - 1, max(0, i));
  j = min(RES - 1, max(0, j));
  k = min(RES - 1, max(0, k));
  return (i * RES + j) * RES + k;
}

#if defined(HAVE_ASYNC_LDS)
typedef __attribute__((address_space(1))) v4i* gas_v4i_p;   // global (AS1)
typedef __attribute__((address_space(3))) v4i* lds_v4i_p;   // LDS (AS3)

// 16B global -> LDS DMA copy (async, tracked by ASYNCcnt).
__device__ __forceinline__ void async_g2l_b128(const float* g, float* l) {
  __builtin_amdgcn_global_load_async_to_lds_b128(
      reinterpret_cast<gas_v4i_p>(reinterpret_cast<uintptr_t>(g)),
      reinterpret_cast<lds_v4i_p>(reinterpret_cast<uintptr_t>(l)),
      0, 0);
}
#endif

// ---------------- pass 0: zero the grid ----------------
__global__ void k_zero_grid(float4* __restrict__ g4) {
  int t = blockIdx.x * blockDim.x + threadIdx.x;   // exactly NCELLS/4 threads
  g4[t] = make_float4(0.f, 0.f, 0.f, 0.f);
}

// ---------------- pass 1: scatter-add importances into grid ----------------
#define SCAT_BLK 256
__global__ void k_scatter(const float* __restrict__ pos,
                          const float* __restrict__ imp,
                          float* __restrict__ grid, int n) {
  __shared__ float s_pos[SCAT_BLK * 3];   // 3 KB
  __shared__ float s_imp[SCAT_BLK];       // 1 KB
  const int t = threadIdx.x;
  for (int base = blockIdx.x * SCAT_BLK; base < n; base += gridDim.x * SCAT_BLK) {
    const int count = min(SCAT_BLK, n - base);
    float px = 0.f, py = 0.f, pz = 0.f, w = 0.f;
#if defined(HAVE_ASYNC_LDS)
    if (count == SCAT_BLK) {
      // Stage 256 points (3072B) + 256 importances (1024B) as coalesced b128 DMA.
      if (t < 192) async_g2l_b128(pos + (size_t)base * 3 + t * 4, s_pos + t * 4);
      if (t < 64)  async_g2l_b128(imp + (size_t)base + t * 4,     s_imp + t * 4);
      __builtin_amdgcn_s_wait_asynccnt(0);
      __syncthreads();
      px = s_pos[3 * t + 0]; py = s_pos[3 * t + 1]; pz = s_pos[3 * t + 2];
      w  = s_imp[t];
      __syncthreads();   // LDS reused next iteration
    } else
#endif
    {
      if (t < count) {
        const int p = base + t;
        px = pos[3 * p + 0]; py = pos[3 * p + 1]; pz = pos[3 * p + 2];
        w  = imp[p];
      }
    }
    if (t < count) {
      atomicAdd(&grid[voxel_index(px, py, pz)], w);   // L2-resident 8MB target
    }
  }
}

// ---------------- pass 2: per-tile sums (WMMA reduction) ----------------
// One block per 2048-cell tile; each wave reduces 256 cells.
__global__ void k_tile_sums(const float* __restrict__ grid,
                            float* __restrict__ partials) {
  const int tile = blockIdx.x;
  const int wave = threadIdx.x >> 5;
  const int lane = threadIdx.x & 31;
  const float* wbase = grid + tile * TILEC + wave * 256;
  float s;
#if defined(HAVE_WMMA_F32X4)
  // D = ones(16x4) * B(4x16) + C : every row of C accumulates colsum(B);
  // sum of all 256 C entries == 16 * sum(B elements), independent of B layout.
  v2f a = {1.0f, 1.0f};
  v8f c = {};
  for (int it = 0; it < 4; ++it) {
    v2f b = *(const v2f*)(wbase + it * 64 + lane * 2);
    c = __builtin_amdgcn_wmma_f32_16x16x4_f32(false, a, false, b,
                                              (short)0, c, false, false);
  }
  s = c[0] + c[1] + c[2] + c[3] + c[4] + c[5] + c[6] + c[7];
  for (int m = 16; m >= 1; m >>= 1) s += __shfl_xor(s, m, 32);
  s *= (1.0f / 16.0f);
#else
  s = 0.f;
  for (int i = lane; i < 256; i += 32) s += wbase[i];
  for (int m = 16; m >= 1; m >>= 1) s += __shfl_xor(s, m, 32);
#endif
  __shared__ float red[8];
  if (lane == 0) red[wave] = s;
  __syncthreads();
  if (threadIdx.x == 0) {
    float tsum = 0.f;
    for (int i = 0; i < 8; ++i) tsum += red[i];
    partials[tile] = tsum;
  }
}

// ---------------- pass 3: exclusive scan of 1024 tile sums (tiny) ----------------
__global__ void k_scan_tiles(float* __restrict__ partials) {
  __shared__ float s[NTILES];
  for (int i = threadIdx.x; i < NTILES; i += blockDim.x) s[i] = partials[i];
  __syncthreads();
  if (threadIdx.x == 0) {
    float acc = 0.f;
    for (int i = 0; i < NTILES; ++i) { float v = s[i]; s[i] = acc; acc += v; }
  }
  __syncthreads();
  for (int i = threadIdx.x; i < NTILES; i += blockDim.x) partials[i] = s[i];
}

// ---------------- pass 4: inclusive CDF per tile ----------------
__global__ void k_cdf(const float* __restrict__ grid,
                      const float* __restrict__ partials,
                      float* __restrict__ cdf) {
  const int tile = blockIdx.x;
  const int tb = tile * TILEC + threadIdx.x * 8;     // 8 cells/thread
  const float4* g4 = (const float4*)(grid + tb);
  float4 x0 = g4[0], x1 = g4[1];
  float v[8] = {x0.x, x0.y, x0.z, x0.w, x1.x, x1.y, x1.z, x1.w};
  float run = 0.f;
  for (int i = 0; i < 8; ++i) { run += v[i]; v[i] = run; }
  __shared__ float s[256];
  s[threadIdx.x] = run;
  __syncthreads();
  for (int off = 1; off < 256; off <<= 1) {            // Hillis-Steele
    float t = (threadIdx.x >= off) ? s[threadIdx.x - off] : 0.f;
    __syncthreads();
    s[threadIdx.x] += t;
    __syncthreads();
  }
  const float offset = partials[tile] + ((threadIdx.x == 0) ? 0.f : s[threadIdx.x - 1]);
  float4* c4 = (float4*)(cdf + tb);
  c4[0] = make_float4(v[0] + offset, v[1] + offset, v[2] + offset, v[3] + offset);
  c4[1] = make_float4(v[4] + offset, v[5] + offset, v[6] + offset, v[7] + offset);
}

// ---------------- pass 5: density + probability for queries ----------------
__global__ void k_gather(const float* __restrict__ qry,
                         const float* __restrict__ grid,
                         const float* __restrict__ cdf,
                         float* __restrict__ dens,
                         float* __restrict__ prob, int m) {
  int q = blockIdx.x * blockDim.x + threadIdx.x;
  if (q >= m) return;
  const float total = fmaxf(cdf[NCELLS - 1], 1e-12f);
  const int idx = voxel_index(qry[3 * q], qry[3 * q + 1], qry[3 * q + 2]);
  const float d = grid[idx];
  dens[q] = d;
  prob[q] = d / total;
}

// ---------------- pass 6: multinomial sampling via CDF binary search ----------------
__global__ void k_sample(const float* __restrict__ cdf,
                         float* __restrict__ out, int nS) {
  int s = blockIdx.x * blockDim.x + threadIdx.x;
  if (s >= nS) return;
  // Deterministic counter-based hash (graph-replay safe).
  unsigned h = (unsigned)s * 0x9E3779B9u + 0x85EBCA6Bu;
  h ^= h >> 16; h *= 0x7FEB352Du;
  h ^= h >> 15; h *= 0x846CA68Bu;
  h ^= h >> 16;
  const float total = cdf[NCELLS - 1];
  const float u = (float)(h >> 8) * (1.0f / 16777216.0f) * total;  // [0,total)
  int lo = 0, hi = NCELLS;        // first idx with cdf[idx] > u  (21 L2 probes)
  while (lo < hi) {
    int mid = (lo + hi) >> 1;
    if (cdf[mid] > u) hi = mid; else lo = mid + 1;
  }
  const int idx = (lo < NCELLS) ? lo : (NCELLS - 1);
  const int i = idx >> 14;
  const int j = (idx >> 7) & (RES - 1);
  const int k = idx & (RES - 1);
  out[3 * s + 0] = ORIGINF + ((float)i + 0.5f) * CELLSZ;
  out[3 * s + 1] = ORIGINF + ((float)j + 0.5f) * CELLSZ;
  out[3 * s + 2] = ORIGINF + ((float)k + 0.5f) * CELLSZ;
}

// ---------------- launcher ----------------
extern "C" void kernel_launch(void* const* d_in, const int* in_sizes, int n_in,
                              void* d_out, int out_size, void* d_ws, size_t ws_size,
                              hipStream_t stream) {
  const float* pos = (const float*)d_in[0];
  const float* imp = (const float*)d_in[1];
  const float* qry = (const float*)d_in[2];
  const int nP = in_sizes[0] / 3;              // 8,000,000
  const int nQ = in_sizes[2] / 3;              // 2,000,000
  const int nS = (out_size - 2 * nQ) / 3;      // 1,000,000

  // workspace layout (floats): grid | tile partials | pad | cdf
  float* ws       = (float*)d_ws;
  float* grid     = ws;
  float* partials = ws + NCELLS;
  float* cdf      = ws + NCELLS + NTILES + 64; // 16B-aligned region

  float* dens = (float*)d_out;
  float* prob = dens + nQ;
  float* smpl = prob + nQ;

  k_zero_grid <<<NCELLS / (4 * 256), 256, 0, stream>>>((float4*)grid);
  k_scatter   <<<2048, SCAT_BLK, 0, stream>>>(pos, imp, grid, nP);
  k_tile_sums <<<NTILES, 256, 0, stream>>>(grid, partials);
  k_scan_tiles<<<1, 256, 0, stream>>>(partials);
  k_cdf       <<<NTILES, 256, 0, stream>>>(grid, partials, cdf);
  k_gather    <<<(nQ + 255) / 256, 256, 0, stream>>>(qry, grid, cdf, dens, prob, nQ);
  k_sample    <<<(nS + 255) / 256, 256, 0, stream>>>(cdf, smpl, nS);
}